// LightGCN_70111046140351
// MI455X (gfx1250) — compile-verified
//
#include <hip/hip_runtime.h>
#include <hip/hip_bf16.h>

#define N_USERS 100000
#define N_ITEMS 200000
#define N_NODES (N_USERS + N_ITEMS)
#define NNZ     4800000
#define DIM     64
#define BATCH   16384

typedef float v2f __attribute__((ext_vector_type(2)));
typedef float v8f __attribute__((ext_vector_type(8)));

// ---------------------------------------------------------------------------
// 1) init: h = acc = concat(user_emb, item_emb), float4 vectorized
// ---------------------------------------------------------------------------
__global__ void lgcn_init(const float* __restrict__ user_emb,
                          const float* __restrict__ item_emb,
                          float* __restrict__ h,
                          float* __restrict__ acc) {
    size_t f = (size_t)blockIdx.x * blockDim.x + threadIdx.x;   // float4 index
    size_t total = (size_t)N_NODES * (DIM / 4);
    if (f >= total) return;
    size_t node = f >> 4;              // DIM/4 == 16 float4 per node
    size_t part = f & 15;
    const float4* src;
    if (node < N_USERS)
        src = (const float4*)(user_emb + node * DIM);
    else
        src = (const float4*)(item_emb + (node - N_USERS) * DIM);
    float4 v = src[part];
    ((float4*)h)[f]   = v;
    ((float4*)acc)[f] = v;
}

// ---------------------------------------------------------------------------
// 2) zero a buffer (float4)
// ---------------------------------------------------------------------------
__global__ void lgcn_zero(float* __restrict__ p) {
    size_t f = (size_t)blockIdx.x * blockDim.x + threadIdx.x;
    size_t total = (size_t)N_NODES * (DIM / 4);
    if (f >= total) return;
    ((float4*)p)[f] = make_float4(0.f, 0.f, 0.f, 0.f);
}

// ---------------------------------------------------------------------------
// 3) SpMM scatter: 16 lanes cooperate per nnz; float4 gather (coalesced 256B
//    per nnz group, L2-resident), hardware f32 atomics for the scatter.
// ---------------------------------------------------------------------------
__global__ void lgcn_spmm(const int*   __restrict__ rows,
                          const int*   __restrict__ cols,
                          const float* __restrict__ vals,
                          const float* __restrict__ h,
                          float*       __restrict__ out) {
    size_t tid = (size_t)blockIdx.x * blockDim.x + threadIdx.x;
    size_t e   = tid >> 4;           // nnz index
    int    part = (int)(tid & 15);   // which float4 chunk of the 64-dim row
    if (e >= (size_t)NNZ) return;
    int   r = rows[e];
    int   c = cols[e];
    float v = vals[e];
    const float4 src = *(const float4*)(h + (size_t)c * DIM + part * 4);
    float* dst = out + (size_t)r * DIM + part * 4;
    unsafeAtomicAdd(dst + 0, v * src.x);
    unsafeAtomicAdd(dst + 1, v * src.y);
    unsafeAtomicAdd(dst + 2, v * src.z);
    unsafeAtomicAdd(dst + 3, v * src.w);
}

// ---------------------------------------------------------------------------
// 4) acc += h  (float4)
// ---------------------------------------------------------------------------
__global__ void lgcn_addacc(float* __restrict__ acc, const float* __restrict__ h) {
    size_t f = (size_t)blockIdx.x * blockDim.x + threadIdx.x;
    size_t total = (size_t)N_NODES * (DIM / 4);
    if (f >= total) return;
    float4 a = ((float4*)acc)[f];
    float4 b = ((const float4*)h)[f];
    a.x += b.x; a.y += b.y; a.z += b.z; a.w += b.w;
    ((float4*)acc)[f] = a;
}

// ---------------------------------------------------------------------------
// 5) Batched dot via V_WMMA_F32_16X16X4_F32.
//    One wave32 handles 16 (user,item) pairs: A = U tile (16x4 chunk),
//    B = V^T tile (4x16 chunk); 16 WMMAs accumulate over K=64; diagonal of
//    the 16x16 product holds the 16 dot products. Grid exactly covers all
//    1024 tiles so EXEC is all-ones at every WMMA.
// ---------------------------------------------------------------------------
__global__ void lgcn_dot_wmma(const float* __restrict__ acc,
                              const int*   __restrict__ users,
                              const int*   __restrict__ items,
                              float*       __restrict__ out) {
    __shared__ float lds[8][16][17];            // 8 waves/block, padded 16x16
    int lane  = threadIdx.x & 31;
    int wv    = threadIdx.x >> 5;               // wave within block (0..7)
    int tile  = blockIdx.x * 8 + wv;            // 0..1023
    int m     = lane & 15;                      // A row / B col index
    int khalf = lane >> 4;                      // 0 or 1

    int u_row = users[tile * 16 + m];
    int v_row = items[tile * 16 + m];
    const float* up = acc + (size_t)u_row * DIM;
    const float* vp = acc + ((size_t)N_USERS + (size_t)v_row) * DIM;

    v8f c = {};
#pragma unroll
    for (int kc = 0; kc < DIM / 4; ++kc) {
        int off = kc * 4 + khalf * 2;
        v2f a; a.x = up[off]; a.y = up[off + 1];
        v2f b; b.x = vp[off]; b.y = vp[off + 1];
        // D = A(16x4) x B(4x16) + C ; f32 throughout
        c = __builtin_amdgcn_wmma_f32_16x16x4_f32(
                /*neg_a=*/false, a, /*neg_b=*/false, b,
                /*c_mod=*/(short)0, c, /*reuse_a=*/false, /*reuse_b=*/false);
    }

    // C layout: VGPR r, lane L -> (M = r + 8*(L>>4), N = L&15). Dump to LDS.
#pragma unroll
    for (int r = 0; r < 8; ++r)
        lds[wv][r + 8 * khalf][m] = c[r];
    __syncthreads();

    // diag(U V^T)[m] = dot(u_m, v_m); fold in the (1/4)*(1/4) averaging.
    if (lane < 16)
        out[tile * 16 + lane] = lds[wv][lane][lane] * (1.0f / 16.0f);
}

// ---------------------------------------------------------------------------
extern "C" void kernel_launch(void* const* d_in, const int* in_sizes, int n_in,
                              void* d_out, int out_size, void* d_ws, size_t ws_size,
                              hipStream_t stream) {
    const float* user_emb = (const float*)d_in[0];
    const float* item_emb = (const float*)d_in[1];
    const int*   g_rows   = (const int*)  d_in[2];
    const int*   g_cols   = (const int*)  d_in[3];
    const float* g_vals   = (const float*)d_in[4];
    const int*   users    = (const int*)  d_in[5];
    const int*   items    = (const int*)  d_in[6];
    float*       out      = (float*)d_out;

    const size_t node_f = (size_t)N_NODES * DIM;     // floats per buffer
    float* bufA = (float*)d_ws;                      // h ping
    float* bufB = bufA + node_f;                     // h pong
    float* accb = bufB + node_f;                     // accumulator

    const int TB = 256;
    const int elem_blocks = (int)(((size_t)N_NODES * (DIM / 4) + TB - 1) / TB);
    const int spmm_blocks = (int)(((size_t)NNZ * 16 + TB - 1) / TB);

    // h = acc = all_ego
    lgcn_init<<<elem_blocks, TB, 0, stream>>>(user_emb, item_emb, bufA, accb);

    float* hsrc = bufA;
    float* hdst = bufB;
    for (int hop = 0; hop < 3; ++hop) {
        lgcn_zero<<<elem_blocks, TB, 0, stream>>>(hdst);
        lgcn_spmm<<<spmm_blocks, TB, 0, stream>>>(g_rows, g_cols, g_vals, hsrc, hdst);
        lgcn_addacc<<<elem_blocks, TB, 0, stream>>>(accb, hdst);
        float* t = hsrc; hsrc = hdst; hdst = t;      // swap ping-pong
    }

    // 1024 tiles of 16 pairs; 8 waves per 256-thread block -> 128 blocks
    lgcn_dot_wmma<<<BATCH / 16 / 8, TB, 0, stream>>>(accb, users, items, out);
}